// GNN_MLP_29566554866533
// MI455X (gfx1250) — compile-verified
//
#include <hip/hip_runtime.h>

typedef float v2f __attribute__((ext_vector_type(2)));
typedef float v8f __attribute__((ext_vector_type(8)));

// ---------------------------------------------------------------------------
// Degree / normalization kernels
// ---------------------------------------------------------------------------
__global__ void k_init_deg(float* __restrict__ deg, int n) {
    int i = blockIdx.x * blockDim.x + threadIdx.x;
    if (i < n) deg[i] = 1.0f;  // self-loop contributes 1 to every node's degree
}

__global__ void k_deg_accum(const long long* __restrict__ dst, float* __restrict__ deg, int E) {
    int e = blockIdx.x * blockDim.x + threadIdx.x;
    if (e < E) atomicAdd(&deg[(int)dst[e]], 1.0f);
}

__global__ void k_dinv(float* __restrict__ deg, int n) {
    int i = blockIdx.x * blockDim.x + threadIdx.x;
    if (i < n) {
        float d = deg[i];
        deg[i] = (d > 0.0f) ? rsqrtf(d) : 0.0f;  // in-place: deg -> deg^{-1/2}
    }
}

// ---------------------------------------------------------------------------
// FP32 WMMA GEMM: D[M,N] = act(A[M,K] * B[K,N] + bias[N])
// One wave computes a 32x32 supertile = 2x2 grid of 16x16 WMMA tiles
// (V_WMMA_F32_16X16X4_F32). A frags reused across both N-tiles, B frags
// across both M-tiles: 6 VMEM ops per 4 WMMAs instead of 3 per 1.
// Requires M,N % 32 == 0, K % 4 == 0 (true for all three GEMMs here).
// A-frag (16x4 f32): lane l<16 holds row m0+l, K = k+2*(lane>>4)+{0,1}.
// B-frag (4x16 f32): lane holds col n0+(lane&15), rows K = k+2*(lane>>4)+{0,1}.
// C/D (16x16 f32):   VGPR r -> row m0 + r + 8*(lane>>4), col n0 + (lane&15).
// ---------------------------------------------------------------------------
__global__ __launch_bounds__(256) void k_gemm_f32_wmma(
    const float* __restrict__ A, const float* __restrict__ B,
    const float* __restrict__ bias, float* __restrict__ D,
    int M, int N, int K, int relu) {
    const int wave = threadIdx.x >> 5;
    const int lane = threadIdx.x & 31;
    const int tN32 = N >> 5;  // 32-wide supertiles along N
    const int tM32 = M >> 5;  // 32-tall supertiles along M

    long tile = (long)blockIdx.x * (blockDim.x >> 5) + wave;
    int tm = (int)(tile / tN32);
    int tn = (int)(tile % tN32);
    if (tm >= tM32) return;  // wave-uniform: EXEC stays all-1s for WMMA

    const int m0 = tm << 5;
    const int n0 = tn << 5;
    const int l  = lane & 15;
    const int hh = lane >> 4;

    const float* __restrict__ A0 = A + (size_t)(m0 + l) * K + 2 * hh;        // rows m0..m0+15
    const float* __restrict__ A1 = A0 + (size_t)16 * K;                      // rows m0+16..m0+31
    const float* __restrict__ B0 = B + (size_t)(2 * hh) * N + n0 + l;        // cols n0..n0+15
    const float* __restrict__ B1 = B0 + 16;                                  // cols n0+16..n0+31

    v8f c00 = {}, c01 = {}, c10 = {}, c11 = {};
    for (int k = 0; k < K; k += 4) {
        v2f a0, a1, b0, b1;
        a0.x = A0[0];         a0.y = A0[1];
        a1.x = A1[0];         a1.y = A1[1];
        b0.x = B0[0];         b0.y = B0[(size_t)N];
        b1.x = B1[0];         b1.y = B1[(size_t)N];
        c00 = __builtin_amdgcn_wmma_f32_16x16x4_f32(false, a0, false, b0, (short)0, c00, false, false);
        c01 = __builtin_amdgcn_wmma_f32_16x16x4_f32(false, a0, false, b1, (short)0, c01, false, false);
        c10 = __builtin_amdgcn_wmma_f32_16x16x4_f32(false, a1, false, b0, (short)0, c10, false, false);
        c11 = __builtin_amdgcn_wmma_f32_16x16x4_f32(false, a1, false, b1, (short)0, c11, false, false);
        A0 += 4;
        A1 += 4;
        B0 += (size_t)4 * N;
        B1 += (size_t)4 * N;
    }

    float bv0 = bias ? bias[n0 + l]      : 0.0f;
    float bv1 = bias ? bias[n0 + 16 + l] : 0.0f;
    float* __restrict__ Dp = D + (size_t)(m0 + 8 * hh) * N + n0 + l;  // top half rows
#pragma unroll
    for (int r = 0; r < 8; ++r) {
        float v00 = c00[r] + bv0;
        float v01 = c01[r] + bv1;
        float v10 = c10[r] + bv0;
        float v11 = c11[r] + bv1;
        if (relu) {
            v00 = fmaxf(v00, 0.0f);
            v01 = fmaxf(v01, 0.0f);
            v10 = fmaxf(v10, 0.0f);
            v11 = fmaxf(v11, 0.0f);
        }
        Dp[(size_t)r * N]                    = v00;  // row m0+8*hh+r,    col n0+l
        Dp[(size_t)r * N + 16]               = v01;  // row m0+8*hh+r,    col n0+16+l
        Dp[(size_t)(r + 16) * N]             = v10;  // row m0+16+8*hh+r, col n0+l
        Dp[(size_t)(r + 16) * N + 16]        = v11;  // row m0+16+8*hh+r, col n0+16+l
    }
}

// ---------------------------------------------------------------------------
// GCN propagation
// ---------------------------------------------------------------------------
// g[i,:] = h[i,:] * dinv[i]^2  (self-loop term; also initializes accumulator)
__global__ void k_self_loop(const float* __restrict__ h, const float* __restrict__ dinv,
                            float* __restrict__ g, size_t total) {
    size_t idx = (size_t)blockIdx.x * blockDim.x + threadIdx.x;
    if (idx >= total) return;
    int row  = (int)(idx >> 7);           // 128 features per row
    float dv = dinv[row];
    g[idx] = h[idx] * dv * dv;
}

// One wave per edge: float4 gather of h[src] (L2-resident) + 4 fp32 atomics.
__global__ __launch_bounds__(256) void k_scatter(
    const long long* __restrict__ src, const long long* __restrict__ dst,
    const float* __restrict__ dinv, const float* __restrict__ h,
    float* __restrict__ g, int E) {
    int e    = blockIdx.x * (blockDim.x >> 5) + (threadIdx.x >> 5);
    int lane = threadIdx.x & 31;
    if (e >= E) return;
    int s = (int)src[e];
    int d = (int)dst[e];
    float norm = dinv[s] * dinv[d];
    const float4 v = ((const float4*)(h + (size_t)s * 128))[lane];
    float* gd = g + (size_t)d * 128 + lane * 4;
    atomicAdd(gd + 0, v.x * norm);
    atomicAdd(gd + 1, v.y * norm);
    atomicAdd(gd + 2, v.z * norm);
    atomicAdd(gd + 3, v.w * norm);
}

// g = relu(g + b_gcn[col]), col = idx & 127
__global__ void k_bias_relu128(float* __restrict__ g, const float* __restrict__ b, size_t total) {
    size_t idx = (size_t)blockIdx.x * blockDim.x + threadIdx.x;
    if (idx >= total) return;
    g[idx] = fmaxf(g[idx] + b[idx & 127], 0.0f);
}

// ---------------------------------------------------------------------------
// log_softmax over 64 logits per row; one wave per row, 2 elements per lane.
// ---------------------------------------------------------------------------
__global__ __launch_bounds__(256) void k_logsoftmax64(
    const float* __restrict__ logits, float* __restrict__ out, int M) {
    int row  = blockIdx.x * (blockDim.x >> 5) + (threadIdx.x >> 5);
    int lane = threadIdx.x & 31;
    if (row >= M) return;
    const float* lp = logits + (size_t)row * 64;
    float v0 = lp[lane];
    float v1 = lp[lane + 32];
    float mx = fmaxf(v0, v1);
#pragma unroll
    for (int off = 16; off; off >>= 1) mx = fmaxf(mx, __shfl_xor(mx, off, 32));
    float s = expf(v0 - mx) + expf(v1 - mx);
#pragma unroll
    for (int off = 16; off; off >>= 1) s += __shfl_xor(s, off, 32);
    float lse = mx + logf(s);
    float* op = out + (size_t)row * 64;
    op[lane]      = v0 - lse;
    op[lane + 32] = v1 - lse;
}

// ---------------------------------------------------------------------------
// Host-side launch
// ---------------------------------------------------------------------------
extern "C" void kernel_launch(void* const* d_in, const int* in_sizes, int n_in,
                              void* d_out, int out_size, void* d_ws, size_t ws_size,
                              hipStream_t stream) {
    const float*     x   = (const float*)d_in[0];
    const long long* ei  = (const long long*)d_in[1];  // int64 edge_index, shape (2,E)
    const float*     Wg  = (const float*)d_in[2];
    const float*     bg  = (const float*)d_in[3];
    const float*     W1  = (const float*)d_in[4];
    const float*     b1  = (const float*)d_in[5];
    const float*     W2  = (const float*)d_in[6];
    const float*     b2  = (const float*)d_in[7];
    float*           out = (float*)d_out;

    const int nN = in_sizes[0] / 128;   // 100000 nodes
    const int E  = in_sizes[1] / 2;     // 800000 edges
    const long long* src = ei;          // edge_index[0]
    const long long* dst = ei + E;      // edge_index[1]

    // Workspace layout (floats):
    //  [0)            dinv      : nN
    //  [100352)       h         : nN*128   (reused later for logits: nN*64)
    //  [+nN*128)      g         : nN*128
    //  [+nN*128)      m         : nN*512
    float* ws     = (float*)d_ws;
    float* dinv   = ws;
    float* h      = ws + 100352;
    float* g      = h + (size_t)nN * 128;
    float* m      = g + (size_t)nN * 128;
    float* logits = h;  // h dead after propagation; 64 <= 128 cols fits

    const int B = 256;

    // 1) degree (with self-loops) -> deg^{-1/2}
    k_init_deg <<<(nN + B - 1) / B, B, 0, stream>>>(dinv, nN);
    k_deg_accum<<<(E  + B - 1) / B, B, 0, stream>>>(dst, dinv, E);
    k_dinv     <<<(nN + B - 1) / B, B, 0, stream>>>(dinv, nN);

    // 2) h = x @ W_gcn   (M=nN, K=128, N=128)  — fp32 WMMA, 32x32/wave
    {
        long tiles  = (long)(nN / 32) * (128 / 32);
        int  blocks = (int)((tiles + 7) / 8);  // 8 waves per 256-thread block
        k_gemm_f32_wmma<<<blocks, B, 0, stream>>>(x, Wg, nullptr, h, nN, 128, 128, 0);
    }

    // 3) propagation: g = dinv^2 * h (self loop) + sum_e norm_e * h[src]
    {
        size_t total = (size_t)nN * 128;
        k_self_loop<<<(unsigned)((total + B - 1) / B), B, 0, stream>>>(h, dinv, g, total);
        int blocks = (E + 7) / 8;  // one wave per edge, 8 edges per block
        k_scatter<<<blocks, B, 0, stream>>>(src, dst, dinv, h, g, E);
        k_bias_relu128<<<(unsigned)((total + B - 1) / B), B, 0, stream>>>(g, bg, total);
    }

    // 4) m = relu(g @ W1 + b1)   (M=nN, K=128, N=512)
    {
        long tiles  = (long)(nN / 32) * (512 / 32);
        int  blocks = (int)((tiles + 7) / 8);
        k_gemm_f32_wmma<<<blocks, B, 0, stream>>>(g, W1, b1, m, nN, 512, 128, 1);
    }

    // 5) logits = m @ W2 + b2    (M=nN, K=512, N=64)
    {
        long tiles  = (long)(nN / 32) * (64 / 32);
        int  blocks = (int)((tiles + 7) / 8);
        k_gemm_f32_wmma<<<blocks, B, 0, stream>>>(m, W2, b2, logits, nN, 64, 512, 0);
    }

    // 6) out = log_softmax(logits) over 64 classes per row
    {
        int blocks = (nN + 7) / 8;  // one wave per row
        k_logsoftmax64<<<blocks, B, 0, stream>>>(logits, out, nN);
    }
}